// DiscoBertModel_5042291605565
// MI455X (gfx1250) — compile-verified
//
#include <hip/hip_runtime.h>
#include <hip/hip_bf16.h>
#include <math.h>

typedef __attribute__((ext_vector_type(2))) float v2f;
typedef __attribute__((ext_vector_type(4))) float v4f;
typedef __attribute__((ext_vector_type(8))) float v8f;

#define HID   200
#define HALF  100
#define NEDUS 32768
#define NSTEPS (2 * NEDUS - 1)      // 65535
#define KPROJ 768
#define FEAT  600                   // 3*HID
#define GATES 500                   // 5*HALF

// ---------------- workspace layout (in floats) ----------------
#define OFF_ENC   ((size_t)0)                         // NEDUS*HID
#define OFF_M     (OFF_ENC + (size_t)NEDUS * HID)     // NEDUS*HID
#define OFF_WCAT  (OFF_M   + (size_t)NEDUS * HID)     // 32*600
#define OFF_BCAT  (OFF_WCAT + (size_t)32 * FEAT)      // 32
#define OFF_PART  (OFF_BCAT + (size_t)32)             // 512 block partials

__device__ __forceinline__ float sigm(float x) { return 1.0f / (1.0f + expf(-x)); }

// =================================================================
// Kernel 0: prep — pack [W_act;W_lab;W_dir] into a zero-padded 32x600
// matrix + 32-entry bias for the WMMA loss GEMM.
// =================================================================
__global__ void prep_kernel(const float* __restrict__ W_act, const float* __restrict__ b_act,
                            const float* __restrict__ W_lab, const float* __restrict__ b_lab,
                            const float* __restrict__ W_dir, const float* __restrict__ b_dir,
                            float* __restrict__ Wcat, float* __restrict__ bcat) {
  int idx = blockIdx.x * blockDim.x + threadIdx.x;
  if (idx < 32 * FEAT) {
    int r = idx / FEAT, c = idx % FEAT;
    float v = 0.0f;
    if (r < 2)       v = W_act[r * FEAT + c];
    else if (r < 21) v = W_lab[(r - 2) * FEAT + c];
    else if (r < 24) v = W_dir[(r - 21) * FEAT + c];
    Wcat[idx] = v;
  }
  if (idx < 32) {
    float v = 0.0f;
    if (idx < 2)       v = b_act[idx];
    else if (idx < 21) v = b_lab[idx - 2];
    else if (idx < 24) v = b_dir[idx - 21];
    bcat[idx] = v;
  }
}

// =================================================================
// Kernel 1: projection GEMM  enc = enc_cls @ W_proj^T + b_proj
// via V_WMMA_F32_16X16X4_F32. One wave computes a 16x32 C slab
// (two 16x16 N tiles sharing each A load -> halves A traffic).
// A layout (16x4 f32): lanes 0-15 hold row M=lane, K={k0,k0+1};
// lanes 16-31 hold K={k0+2,k0+3}. B mirrored with N on lanes.
// Grid: 2048 M-tiles x 7 N-slabs = 14336 slabs, 8 waves/block.
// =================================================================
__global__ void proj_wmma_kernel(const float* __restrict__ A,     // 32768x768
                                 const float* __restrict__ W,     // 200x768
                                 const float* __restrict__ bias,  // 200
                                 float* __restrict__ C) {         // 32768x200
  const int lane = threadIdx.x & 31;
  const int wave = threadIdx.x >> 5;
  const int NSLAB = 7;                               // ceil(200/32)
  int slab = blockIdx.x * (blockDim.x >> 5) + wave;
  int mt = slab / NSLAB, ns = slab % NSLAB;
  int m0 = mt * 16, n0 = ns * 32;
  int lr = lane & 15, hi = lane >> 4;

  const float* aptr = A + (size_t)(m0 + lr) * KPROJ + 2 * hi;
  int nc0 = n0 + lr;
  int nc1 = n0 + 16 + lr;
  bool v0 = (nc0 < HID), v1 = (nc1 < HID);
  // padded columns read W row 0 (finite garbage); their C columns are
  // never stored, and WMMA columns do not cross-contaminate.
  const float* bp0 = W + (size_t)(v0 ? nc0 : 0) * KPROJ + 2 * hi;
  const float* bp1 = W + (size_t)(v1 ? nc1 : 0) * KPROJ + 2 * hi;

  v8f acc0 = {}, acc1 = {};
#pragma unroll 4
  for (int k = 0; k < KPROJ; k += 4) {
    v2f a  = *(const v2f*)(aptr + k);
    v2f b0 = *(const v2f*)(bp0 + k);
    v2f b1 = *(const v2f*)(bp1 + k);
    acc0 = __builtin_amdgcn_wmma_f32_16x16x4_f32(false, a, false, b0,
                                                 (short)0, acc0, false, false);
    acc1 = __builtin_amdgcn_wmma_f32_16x16x4_f32(false, a, false, b1,
                                                 (short)0, acc1, false, false);
  }
  // C/D layout: VGPR v -> row m0 + v + 8*hi, col = tile base + lr
  if (v0) {
    float bv = bias[nc0];
#pragma unroll
    for (int v = 0; v < 8; ++v)
      C[(size_t)(m0 + v + 8 * hi) * HID + nc0] = acc0[v] + bv;
  }
  if (v1) {
    float bv = bias[nc1];
#pragma unroll
    for (int v = 0; v < 8; ++v)
      C[(size_t)(m0 + v + 8 * hi) * HID + nc1] = acc1[v] + bv;
  }
}

// =================================================================
// Kernel 2: sequential treelstm chain (single resident workgroup).
// m_0 = enc[0]; m_k = treelstm(m_{k-1}, enc[k]), k = 1..N-1.
// 512 threads: 500 compute one gate row each via b128 reads of the
// row-major W_tree (L2-resident, 400KB) and b128 LDS broadcasts of
// the 200-dim state; 100 threads apply the activations.
// =================================================================
__global__ void chain_kernel(const float* __restrict__ enc,     // NEDUS x HID
                             float* __restrict__ m,             // NEDUS x HID
                             const float* __restrict__ W_tree,  // GATES x HID
                             const float* __restrict__ b_tree) {
  __shared__ __align__(16) float x[2 * HALF];      // [h1 ; h2]
  __shared__ float c1s[HALF], c2s[HALF];
  __shared__ float g[GATES];
  const int tid = threadIdx.x;

  if (tid < HID) m[tid] = enc[tid];                 // m_0 = enc[0]
  if (tid < HALF) {
    x[tid]        = enc[tid];                       // h1 = enc[0][:100]
    c1s[tid]      = enc[HALF + tid];                // c1 = enc[0][100:]
    x[HALF + tid] = enc[HID + tid];                 // h2 = enc[1][:100]
    c2s[tid]      = enc[HID + HALF + tid];          // c2 = enc[1][100:]
  }
  const v4f* wrow = (const v4f*)(W_tree + (size_t)(tid < GATES ? tid : 0) * HID);
  __syncthreads();

  for (int k = 1; k < NEDUS; ++k) {
    if (tid < GATES) {
      float acc = b_tree[tid];
#pragma unroll 10
      for (int j4 = 0; j4 < HID / 4; ++j4) {
        v4f w  = wrow[j4];                           // global_load_b128
        v4f xv = *(const v4f*)(x + 4 * j4);          // ds_load_b128 broadcast
        acc = fmaf(w.x, xv.x, acc);
        acc = fmaf(w.y, xv.y, acc);
        acc = fmaf(w.z, xv.z, acc);
        acc = fmaf(w.w, xv.w, acc);
      }
      g[tid] = acc;
    }
    if (tid < 8 && (k + 2) < NEDUS)                  // warm caches for next leaf
      __builtin_prefetch(enc + (size_t)(k + 2) * HID + tid * 25, 0, 1);
    __syncthreads();
    if (tid < HALF) {
      float gi = g[tid], gf1 = g[HALF + tid], gf2 = g[2 * HALF + tid];
      float go = g[3 * HALF + tid], gu = g[4 * HALF + tid];
      float c = sigm(gi) * tanhf(gu) + sigm(gf1) * c1s[tid] + sigm(gf2) * c2s[tid];
      float h = sigm(go) * tanhf(c);
      m[(size_t)k * HID + tid]        = h;
      m[(size_t)k * HID + HALF + tid] = c;
      x[tid] = h;  c1s[tid] = c;                    // e1 <- m_k
      if (k + 1 < NEDUS) {                          // e2 <- enc[k+1]
        x[HALF + tid] = enc[(size_t)(k + 1) * HID + tid];
        c2s[tid]      = enc[(size_t)(k + 1) * HID + HALF + tid];
      }
    }
    __syncthreads();
  }
}

// =================================================================
// Kernel 3: loss GEMM + CE. logits(65535x24) = feats @ Wcat^T, with
// feats(t) = [s1;s0;b] gathered from {missing, enc, m} by parity of t.
// One wave per 16-step M tile; two 16-wide N tiles share one A load.
// =================================================================
__device__ __forceinline__ float feat_elem(const float* p0, const float* p1,
                                           const float* p2, int j) {
  const float* p = (j < HID) ? p0 : ((j < 2 * HID) ? p1 : p2);
  int off = (j < HID) ? j : ((j < 2 * HID) ? (j - HID) : (j - 2 * HID));
  return p[off];
}

__global__ void loss_wmma_kernel(const float* __restrict__ enc,
                                 const float* __restrict__ m,
                                 const float* __restrict__ miss,
                                 const float* __restrict__ Wcat,  // 32x600
                                 const float* __restrict__ bcat,  // 32
                                 const int* __restrict__ ga, const int* __restrict__ gl,
                                 const int* __restrict__ gd,
                                 float* __restrict__ partials) {
  __shared__ float lg[8][16][33];      // [wave][row][logit], padded stride
  __shared__ float lossbuf[256];
  const int lane = threadIdx.x & 31;
  const int wave = threadIdx.x >> 5;
  const int tile = blockIdx.x * 8 + wave;          // 4096 tiles total
  const int t0 = tile * 16;
  const int lr = lane & 15, hi = lane >> 4;

  // ---- per-lane feature segment base pointers for row t = t0+lr ----
  int t = t0 + lr;
  int tc = (t < NSTEPS) ? t : 0;
  int k2 = tc >> 1;
  bool even = ((tc & 1) == 0);
  const float* p0 = even ? (m + (size_t)(k2 - 1) * HID) : miss;               // s1
  const float* p1 = even ? (enc + (size_t)k2 * HID) : (m + (size_t)k2 * HID); // s0
  const float* p2 = even ? ((k2 + 1 < NEDUS) ? (enc + (size_t)(k2 + 1) * HID) : miss)
                         : (enc + (size_t)(k2 + 1) * HID);                    // b
  if (tc == 0) { p0 = miss; p1 = miss; p2 = enc; }
  if (tc == 1) { p0 = miss; p1 = enc;  p2 = enc + HID; }

  const float* bq0 = Wcat + (size_t)lr * FEAT + 2 * hi;          // N = 0..15
  const float* bq1 = Wcat + (size_t)(16 + lr) * FEAT + 2 * hi;   // N = 16..31

  v8f acc0 = {}, acc1 = {};
#pragma unroll 2
  for (int k = 0; k < FEAT; k += 4) {
    int j0 = k + 2 * hi;
    v2f a;
    a.x = feat_elem(p0, p1, p2, j0);
    a.y = feat_elem(p0, p1, p2, j0 + 1);
    v2f b0 = *(const v2f*)(bq0 + k);
    v2f b1 = *(const v2f*)(bq1 + k);
    acc0 = __builtin_amdgcn_wmma_f32_16x16x4_f32(false, a, false, b0,
                                                 (short)0, acc0, false, false);
    acc1 = __builtin_amdgcn_wmma_f32_16x16x4_f32(false, a, false, b1,
                                                 (short)0, acc1, false, false);
  }
  // spill C tiles to LDS: VGPR v -> row v + 8*hi, col lr (+16 for tile 1)
#pragma unroll
  for (int v = 0; v < 8; ++v) {
    lg[wave][v + 8 * hi][lr]      = acc0[v];
    lg[wave][v + 8 * hi][16 + lr] = acc1[v];
  }
  __syncthreads();

  // ---- per-row cross-entropy (lanes 0-15 each own one step) ----
  float myloss = 0.0f;
  if (lane < 16) {
    int tt = t0 + lane;
    if (tt < NSTEPS) {
      const float* L = &lg[wave][lane][0];
      // action: logits 0..1
      {
        float l0 = L[0] + bcat[0], l1 = L[1] + bcat[1];
        float mx = fmaxf(l0, l1);
        float lse = mx + logf(expf(l0 - mx) + expf(l1 - mx));
        int y = ga[tt];
        myloss += lse - (L[y] + bcat[y]);
      }
      // label: logits 2..20
      {
        float mx = -INFINITY;
        for (int i = 2; i < 21; ++i) mx = fmaxf(mx, L[i] + bcat[i]);
        float s = 0.0f;
        for (int i = 2; i < 21; ++i) s += expf(L[i] + bcat[i] - mx);
        int y = 2 + gl[tt];
        myloss += mx + logf(s) - (L[y] + bcat[y]);
      }
      // direction: logits 21..23
      {
        float mx = -INFINITY;
        for (int i = 21; i < 24; ++i) mx = fmaxf(mx, L[i] + bcat[i]);
        float s = 0.0f;
        for (int i = 21; i < 24; ++i) s += expf(L[i] + bcat[i] - mx);
        int y = 21 + gd[tt];
        myloss += mx + logf(s) - (L[y] + bcat[y]);
      }
    }
  }
  lossbuf[threadIdx.x] = myloss;
  __syncthreads();
  if (threadIdx.x == 0) {                         // fixed-order: deterministic
    float s = 0.0f;
    for (int i = 0; i < 256; ++i) s += lossbuf[i];
    partials[blockIdx.x] = s;
  }
}

// =================================================================
// Kernel 4: deterministic final reduction of 512 block partials.
// =================================================================
__global__ void reduce_kernel(const float* __restrict__ partials, int n,
                              float* __restrict__ out) {
  __shared__ float s[256];
  int tid = threadIdx.x;
  float v = 0.0f;
  for (int i = tid; i < n; i += 256) v += partials[i];
  s[tid] = v;
  __syncthreads();
  for (int w = 128; w > 0; w >>= 1) {
    if (tid < w) s[tid] += s[tid + w];
    __syncthreads();
  }
  if (tid == 0) out[0] = s[0];
}

// =================================================================
extern "C" void kernel_launch(void* const* d_in, const int* in_sizes, int n_in,
                              void* d_out, int out_size, void* d_ws, size_t ws_size,
                              hipStream_t stream) {
  const float* enc_cls  = (const float*)d_in[0];
  const float* W_proj   = (const float*)d_in[1];
  const float* b_proj   = (const float*)d_in[2];
  const float* missing  = (const float*)d_in[3];
  const float* W_act    = (const float*)d_in[4];
  const float* b_act    = (const float*)d_in[5];
  const float* W_lab    = (const float*)d_in[6];
  const float* b_lab    = (const float*)d_in[7];
  const float* W_dir    = (const float*)d_in[8];
  const float* b_dir    = (const float*)d_in[9];
  const float* W_tree   = (const float*)d_in[10];
  const float* b_tree   = (const float*)d_in[11];
  const int*   ga       = (const int*)d_in[12];
  const int*   gl       = (const int*)d_in[13];
  const int*   gd       = (const int*)d_in[14];

  float* ws   = (float*)d_ws;
  float* enc  = ws + OFF_ENC;
  float* m    = ws + OFF_M;
  float* Wcat = ws + OFF_WCAT;
  float* bcat = ws + OFF_BCAT;
  float* part = ws + OFF_PART;

  // 0) prep: 32*600 = 19200 elements
  prep_kernel<<<(32 * FEAT + 255) / 256, 256, 0, stream>>>(
      W_act, b_act, W_lab, b_lab, W_dir, b_dir, Wcat, bcat);

  // 1) projection GEMM: 2048 M-tiles * 7 N-slabs / 8 waves = 1792 blocks
  proj_wmma_kernel<<<1792, 256, 0, stream>>>(enc_cls, W_proj, b_proj, enc);

  // 2) sequential treelstm chain (single workgroup)
  chain_kernel<<<1, 512, 0, stream>>>(enc, m, W_tree, b_tree);

  // 3) loss GEMM + CE: 4096 tiles / 8 waves = 512 blocks
  loss_wmma_kernel<<<512, 256, 0, stream>>>(enc, m, missing, Wcat, bcat,
                                            ga, gl, gd, part);

  // 4) deterministic reduce -> d_out[0]
  reduce_kernel<<<1, 256, 0, stream>>>(part, 512, (float*)d_out);
}